// FastRGCNConv_56573309223583
// MI455X (gfx1250) — compile-verified
//
#include <hip/hip_runtime.h>
#include <hip/hip_bf16.h>

#define N_NODES 50000
#define N_EDGES 640000
#define N_REL   8
#define C       128          // IN_C == OUT_C == 128
#define TILE_M  256          // rows per block (8 waves x 32 rows)

typedef __attribute__((ext_vector_type(2))) float v2f;
typedef __attribute__((ext_vector_type(8))) float v8f;

// ---------------------------------------------------------------------------
// out[n][c] = bias[c]
// ---------------------------------------------------------------------------
__global__ void rgcn_init_out(float* __restrict__ out, const float* __restrict__ bias) {
    int i = blockIdx.x * 256 + threadIdx.x;
    if (i < N_NODES * C) out[i] = bias[i & (C - 1)];
}

// ---------------------------------------------------------------------------
// tmp[0:50000, 0:128] = x @ W    (one relation's 128x128 weight)
// Block: 256 threads = 8 waves. Block tile: 256 rows x 128 cols.
// Wave w owns rows m0..m0+31 (two 16-row WMMA strips sharing every B
// fragment) and all 8 column tiles: 16 x v8f accumulators = 128 VGPRs.
// W is staged in LDS pre-swizzled as float2{W[2j][c], W[2j+1][c]} so each
// B fragment of v_wmma_f32_16x16x4_f32 is one b64 LDS read per lane
// (pairs get merged into ds_load_2addr_b64 by the compiler).
// Per K-step per wave: 2 global b64 (A) + 4 ds_load_2addr_b64 (B) + 16 WMMA.
// ---------------------------------------------------------------------------
__global__ __launch_bounds__(256) void rgcn_gemm(const float* __restrict__ x,
                                                 const float* __restrict__ Wg,
                                                 float* __restrict__ tmp) {
    __shared__ float2 sW[64 * C];   // 64 KB: index (k>>1)*128 + col, .x=k even .y=k odd

    const int t = threadIdx.x;

    // -------- cooperative load + swizzle of W (128x128 f32, coalesced b128) ---
    const float4* W4 = (const float4*)Wg;          // 4096 float4s
    #pragma unroll
    for (int i = 0; i < 16; ++i) {
        int idx = t + i * 256;
        int k   = idx >> 5;                        // row 0..127
        int c4  = idx & 31;                        // float4 column
        float4 v = W4[idx];
        int j = k >> 1, p = k & 1;
        float* dst = (float*)&sW[j * C + c4 * 4];  // 4 consecutive float2 slots
        dst[0 * 2 + p] = v.x;
        dst[1 * 2 + p] = v.y;
        dst[2 * 2 + p] = v.z;
        dst[3 * 2 + p] = v.w;
    }
    __syncthreads();

    const int wave = t >> 5;
    const int lane = t & 31;
    const int half = lane >> 4;                    // 0: K+0/K+1 , 1: K+2/K+3
    const int l    = lane & 15;

    const int m0 = blockIdx.x * TILE_M + wave * 32;
    int r0 = m0 + l;        if (r0 >= N_NODES) r0 = N_NODES - 1;  // clamp tail
    int r1 = m0 + 16 + l;   if (r1 >= N_NODES) r1 = N_NODES - 1;  // (never stored)
    const float* ap0 = x + (size_t)r0 * C + 2 * half;
    const float* ap1 = x + (size_t)r1 * C + 2 * half;

    v8f acc[16];
    #pragma unroll
    for (int n = 0; n < 16; ++n) acc[n] = (v8f){0.f,0.f,0.f,0.f,0.f,0.f,0.f,0.f};

    // -------- K loop: 32 steps of K=4 ----------------------------------------
    #pragma unroll 4
    for (int kk = 0; kk < 32; ++kk) {
        v2f a0, a1;                                // A[m][4kk+2h .. +1]
        a0.x = ap0[kk * 4 + 0];  a0.y = ap0[kk * 4 + 1];
        a1.x = ap1[kk * 4 + 0];  a1.y = ap1[kk * 4 + 1];
        const float2* brow = &sW[(kk * 2 + half) * C + l];
        #pragma unroll
        for (int n = 0; n < 8; ++n) {
            float2 bv = brow[n * 16];              // {W[krow][col], W[krow+1][col]}
            v2f b; b.x = bv.x; b.y = bv.y;
            acc[n] = __builtin_amdgcn_wmma_f32_16x16x4_f32(
                         false, a0, false, b, (short)0, acc[n], false, false);
            acc[n + 8] = __builtin_amdgcn_wmma_f32_16x16x4_f32(
                         false, a1, false, b, (short)0, acc[n + 8], false, false);
        }
    }

    // -------- store C tiles: VGPR v -> row m0 + v + 8*half (+16), col n*16+l --
    #pragma unroll
    for (int n = 0; n < 8; ++n) {
        int col = n * 16 + l;
        #pragma unroll
        for (int v = 0; v < 8; ++v) {
            int row0 = m0 + v + 8 * half;
            int row1 = row0 + 16;
            if (row0 < N_NODES) tmp[(size_t)row0 * C + col] = acc[n][v];
            if (row1 < N_NODES) tmp[(size_t)row1 * C + col] = acc[n + 8][v];
        }
    }
}

// ---------------------------------------------------------------------------
// For every edge e with edge_type[e]==rel:
//   out[row[e], :] += tmp[col[e], :]
// Wave-per-edge grid-stride loop (amortizes dispatch over ~39 edges/wave).
// 4 channels per lane: float4 gather (L2-hot: tmp was just written) +
// 4 global_atomic_add_f32.
// ---------------------------------------------------------------------------
__global__ __launch_bounds__(256) void rgcn_scatter(const float* __restrict__ tmp,
                                                    const int* __restrict__ edge_index,
                                                    const int* __restrict__ edge_type,
                                                    float* __restrict__ out,
                                                    int rel) {
    const int waveId = (blockIdx.x * 256 + threadIdx.x) >> 5;
    const int ln     = threadIdx.x & 31;
    const int nWaves = gridDim.x * 8;
    for (int e = waveId; e < N_EDGES; e += nWaves) {
        if (edge_type[e] != rel) continue;
        int row = edge_index[e];                 // edge_index[0][e]: scatter dest
        int col = edge_index[N_EDGES + e];       // edge_index[1][e]: gather source
        const float4 v = *(const float4*)&tmp[(size_t)col * C + ln * 4];
        float* o = &out[(size_t)row * C + ln * 4];
        unsafeAtomicAdd(o + 0, v.x);
        unsafeAtomicAdd(o + 1, v.y);
        unsafeAtomicAdd(o + 2, v.z);
        unsafeAtomicAdd(o + 3, v.w);
    }
}

// ---------------------------------------------------------------------------
extern "C" void kernel_launch(void* const* d_in, const int* in_sizes, int n_in,
                              void* d_out, int out_size, void* d_ws, size_t ws_size,
                              hipStream_t stream) {
    const float* x          = (const float*)d_in[0];   // [50000,128]
    const int*   edge_index = (const int*)  d_in[1];   // [2,640000]
    const int*   edge_type  = (const int*)  d_in[2];   // [640000]
    const float* weight     = (const float*)d_in[3];   // [8,128,128]
    const float* bias       = (const float*)d_in[4];   // [128]
    float* out = (float*)d_out;                        // [50000,128]
    float* tmp = (float*)d_ws;                         // 50000*128 f32 = 25.6 MB

    rgcn_init_out<<<(N_NODES * C + 255) / 256, 256, 0, stream>>>(out, bias);

    const int gemm_blocks    = (N_NODES + TILE_M - 1) / TILE_M;     // 196
    const int scatter_blocks = 2048;                                // grid-stride

    for (int r = 0; r < N_REL; ++r) {
        rgcn_gemm<<<gemm_blocks, 256, 0, stream>>>(
            x, weight + (size_t)r * C * C, tmp);
        rgcn_scatter<<<scatter_blocks, 256, 0, stream>>>(
            tmp, edge_index, edge_type, out, r);
    }
}